// GConvGRU_Temporal_30459908063368
// MI455X (gfx1250) — compile-verified
//
#include <hip/hip_runtime.h>
#include <hip/hip_bf16.h>

typedef float v2f __attribute__((ext_vector_type(2)));
typedef float v8f __attribute__((ext_vector_type(8)));

#define K_CHEB 5

// ---------------------------------------------------------------------------
// Elementwise helpers
// ---------------------------------------------------------------------------
__global__ void zero_kernel(float* __restrict__ p, int n) {
  int i = blockIdx.x * blockDim.x + threadIdx.x;
  if (i < n) p[i] = 0.0f;
}

__global__ void scale_copy_kernel(float* __restrict__ dst, const float* __restrict__ src,
                                  float alpha, int n) {
  int i = blockIdx.x * blockDim.x + threadIdx.x;
  if (i < n) dst[i] = alpha * src[i];
}

// ---------------------------------------------------------------------------
// Graph normalization: deg -> d^-1/2 -> wn[e] = -dis[row]*w*dis[col]
// ---------------------------------------------------------------------------
__global__ void deg_kernel(const int* __restrict__ row, const float* __restrict__ ew,
                           float* __restrict__ deg, int E) {
  int e = blockIdx.x * blockDim.x + threadIdx.x;
  if (e < E) atomicAdd(&deg[row[e]], ew[e]);
}

__global__ void dis_kernel(const float* __restrict__ deg, float* __restrict__ dis, int N) {
  int i = blockIdx.x * blockDim.x + threadIdx.x;
  if (i < N) {
    float d = deg[i];
    dis[i] = (d > 0.0f) ? rsqrtf(fmaxf(d, 1e-30f)) : 0.0f;
  }
}

__global__ void wn_kernel(const int* __restrict__ row, const int* __restrict__ col,
                          const float* __restrict__ ew, const float* __restrict__ dis,
                          float* __restrict__ wn, int E) {
  int e = blockIdx.x * blockDim.x + threadIdx.x;
  if (e < E) wn[e] = -dis[row[e]] * ew[e] * dis[col[e]];
}

// ---------------------------------------------------------------------------
// SpMM: out[col[e]] += scale * wn[e] * v[row[e]]   (F = 32 features, 1 wave/edge-group)
// ---------------------------------------------------------------------------
__global__ void spmm32_kernel(const int* __restrict__ row, const int* __restrict__ col,
                              const float* __restrict__ wn, const float* __restrict__ v,
                              float* __restrict__ out, float scale, int E) {
  int t = blockIdx.x * blockDim.x + threadIdx.x;
  int e = t >> 5;
  int f = t & 31;
  if (e >= E) return;
  float w = scale * wn[e];
  int r = row[e];
  int c = col[e];
  atomicAdd(&out[(size_t)c * 32 + f], w * v[(size_t)r * 32 + f]);
}

// ---------------------------------------------------------------------------
// WMMA helper: D = A(16x4,f32) * B(4x16,f32) + C(16x16,f32)
// ---------------------------------------------------------------------------
__device__ __forceinline__ v8f wmma4(v2f a, v2f b, v8f c) {
  return __builtin_amdgcn_wmma_f32_16x16x4_f32(false, a, false, b, (short)0, c, false, false);
}

// ---------------------------------------------------------------------------
// Cell GEMM (Fin=32, Fout=32): h1 = relu((1-sigmoid(sum Tk@WZk + bZ)) * tanh(sum Tk@WHk + bH))
// One wave per 16-row node tile; 2 gates x 2 N-tiles share the A fragment.
// ---------------------------------------------------------------------------
__global__ __launch_bounds__(256)
void gemm_cell1_kernel(const float* __restrict__ T0, const float* __restrict__ T1,
                       const float* __restrict__ T2, const float* __restrict__ T3,
                       const float* __restrict__ T4,
                       const float* __restrict__ Wx,   // [3, K, 32, 32]
                       const float* __restrict__ bx,   // [3, 32]
                       const float* __restrict__ bh,   // [3, 32]
                       float* __restrict__ h1, int nTiles) {
  int tile = blockIdx.x * 8 + (threadIdx.x >> 5);
  if (tile >= nTiles) return;
  int l  = threadIdx.x & 31;
  int n0 = l & 15;      // A: M row, B/C/D: N col
  int kh = l >> 4;      // half-wave selects K pair

  const float* Tk[K_CHEB] = {T0, T1, T2, T3, T4};
  const float* WZ = Wx;                          // gate 0 (z)
  const float* WH = Wx + 2 * K_CHEB * 32 * 32;   // gate 2 (h)

  v8f aZ0 = {}, aZ1 = {}, aH0 = {}, aH1 = {};
#pragma unroll
  for (int k = 0; k < K_CHEB; ++k) {
    const float* Arow = Tk[k] + ((size_t)tile * 16 + n0) * 32;
#pragma unroll
    for (int kk = 0; kk < 8; ++kk) {
      int c = kk * 4 + kh * 2;
      v2f a;   a.x = Arow[c];              a.y = Arow[c + 1];
      const float* wz = WZ + k * 1024 + c * 32;
      const float* wh = WH + k * 1024 + c * 32;
      v2f bz0; bz0.x = wz[n0];             bz0.y = wz[32 + n0];
      v2f bz1; bz1.x = wz[16 + n0];        bz1.y = wz[48 + n0];
      v2f bh0; bh0.x = wh[n0];             bh0.y = wh[32 + n0];
      v2f bh1; bh1.x = wh[16 + n0];        bh1.y = wh[48 + n0];
      aZ0 = wmma4(a, bz0, aZ0);
      aZ1 = wmma4(a, bz1, aZ1);
      aH0 = wmma4(a, bh0, aH0);
      aH1 = wmma4(a, bh1, aH1);
    }
  }

  float bZlo = bx[n0]      + bh[n0];
  float bZhi = bx[16 + n0] + bh[16 + n0];
  float bHlo = bx[64 + n0] + bh[64 + n0];
  float bHhi = bx[80 + n0] + bh[80 + n0];

#pragma unroll
  for (int i = 0; i < 8; ++i) {
    int m = i + kh * 8;
    size_t ro = ((size_t)tile * 16 + m) * 32;
    float z  = 1.0f / (1.0f + __expf(-(aZ0[i] + bZlo)));
    float th = tanhf(aH0[i] + bHlo);
    h1[ro + n0] = fmaxf((1.0f - z) * th, 0.0f);
    z  = 1.0f / (1.0f + __expf(-(aZ1[i] + bZhi)));
    th = tanhf(aH1[i] + bHhi);
    h1[ro + 16 + n0] = fmaxf((1.0f - z) * th, 0.0f);
  }
}

// ---------------------------------------------------------------------------
// Cell 2 GEMM (Fin=32, Fout=16)
// ---------------------------------------------------------------------------
__global__ __launch_bounds__(256)
void gemm_cell2_kernel(const float* __restrict__ T0, const float* __restrict__ T1,
                       const float* __restrict__ T2, const float* __restrict__ T3,
                       const float* __restrict__ T4,
                       const float* __restrict__ Wx,   // [3, K, 32, 16]
                       const float* __restrict__ bx,   // [3, 16]
                       const float* __restrict__ bh,   // [3, 16]
                       float* __restrict__ h2, int nTiles) {
  int tile = blockIdx.x * 8 + (threadIdx.x >> 5);
  if (tile >= nTiles) return;
  int l  = threadIdx.x & 31;
  int n0 = l & 15;
  int kh = l >> 4;

  const float* Tk[K_CHEB] = {T0, T1, T2, T3, T4};
  const float* WZ = Wx;
  const float* WH = Wx + 2 * K_CHEB * 32 * 16;

  v8f aZ = {}, aH = {};
#pragma unroll
  for (int k = 0; k < K_CHEB; ++k) {
    const float* Arow = Tk[k] + ((size_t)tile * 16 + n0) * 32;
#pragma unroll
    for (int kk = 0; kk < 8; ++kk) {
      int c = kk * 4 + kh * 2;
      v2f a;  a.x = Arow[c];          a.y = Arow[c + 1];
      const float* wz = WZ + k * 512 + c * 16;
      const float* wh = WH + k * 512 + c * 16;
      v2f bz; bz.x = wz[n0];          bz.y = wz[16 + n0];
      v2f bv; bv.x = wh[n0];          bv.y = wh[16 + n0];
      aZ = wmma4(a, bz, aZ);
      aH = wmma4(a, bv, aH);
    }
  }

  float bZ = bx[n0]      + bh[n0];
  float bH = bx[32 + n0] + bh[32 + n0];

#pragma unroll
  for (int i = 0; i < 8; ++i) {
    int m = i + kh * 8;
    size_t ro = ((size_t)tile * 16 + m) * 16;
    float z  = 1.0f / (1.0f + __expf(-(aZ[i] + bZ)));
    float th = tanhf(aH[i] + bH);
    h2[ro + n0] = fmaxf((1.0f - z) * th, 0.0f);
  }
}

// ---------------------------------------------------------------------------
// Output GEMM: out[N,12] = h2[N,16] @ Wl.T[16,12] + bl  (Wl is [12,16] row-major)
// ---------------------------------------------------------------------------
__global__ __launch_bounds__(256)
void gemm_out_kernel(const float* __restrict__ h2, const float* __restrict__ Wl,
                     const float* __restrict__ bl, float* __restrict__ out, int nTiles) {
  int tile = blockIdx.x * 8 + (threadIdx.x >> 5);
  if (tile >= nTiles) return;
  int l  = threadIdx.x & 31;
  int n0 = l & 15;
  int kh = l >> 4;
  int p  = (n0 < 12) ? n0 : 0;   // clamped index: always-valid loads, select to 0 later

  v8f acc = {};
  const float* Arow = h2 + ((size_t)tile * 16 + n0) * 16;
#pragma unroll
  for (int kk = 0; kk < 4; ++kk) {
    int c = kk * 4 + kh * 2;
    v2f a; a.x = Arow[c]; a.y = Arow[c + 1];
    float w0 = Wl[p * 16 + c];
    float w1 = Wl[p * 16 + c + 1];
    v2f b; b.x = (n0 < 12) ? w0 : 0.0f;
           b.y = (n0 < 12) ? w1 : 0.0f;
    acc = wmma4(a, b, acc);
  }

  if (n0 < 12) {
    float bb = bl[n0];
#pragma unroll
    for (int i = 0; i < 8; ++i) {
      int m = i + kh * 8;
      out[((size_t)tile * 16 + m) * 12 + n0] = acc[i] + bb;
    }
  }
}

// ---------------------------------------------------------------------------
// Launcher
// ---------------------------------------------------------------------------
extern "C" void kernel_launch(void* const* d_in, const int* in_sizes, int n_in,
                              void* d_out, int out_size, void* d_ws, size_t ws_size,
                              hipStream_t stream) {
  const float* x   = (const float*)d_in[0];
  const int*   ei  = (const int*)d_in[1];
  const float* ew  = (const float*)d_in[2];
  const float* Wx1 = (const float*)d_in[3];
  const float* bx1 = (const float*)d_in[4];
  const float* bh1 = (const float*)d_in[6];   // Wh1 (d_in[5]) unused: H=0
  const float* Wx2 = (const float*)d_in[7];
  const float* bx2 = (const float*)d_in[8];
  const float* bh2 = (const float*)d_in[10];  // Wh2 (d_in[9]) unused: H=0
  const float* Wl  = (const float*)d_in[11];
  const float* bl  = (const float*)d_in[12];
  float* out = (float*)d_out;

  const int E = in_sizes[2];
  const int N = in_sizes[0] / 32;
  const int* row = ei;
  const int* col = ei + E;
  const int nf = N * 32;
  const int nTiles = N / 16;

  // Workspace layout (floats)
  float* ws  = (float*)d_ws;
  float* deg = ws;
  float* dis = deg + N;
  float* wn  = dis + N;
  float* T1  = wn + E;
  float* T2  = T1 + (size_t)nf;
  float* T3  = T2 + (size_t)nf;
  float* T4  = T3 + (size_t)nf;
  float* h1  = T4 + (size_t)nf;
  float* h2  = h1 + (size_t)nf;

  auto cdiv = [](int a, int b) { return (a + b - 1) / b; };
  const int B = 256;
  const int gN  = cdiv(N, B);
  const int gE  = cdiv(E, B);
  const int gF  = cdiv(nf, B);
  const int gEF = cdiv(E * 32, B);   // 51.2M threads -> fits int
  const int gT  = cdiv(nTiles, 8);

  // 1) normalization weights
  zero_kernel<<<gN, B, 0, stream>>>(deg, N);
  deg_kernel<<<gE, B, 0, stream>>>(row, ew, deg, E);
  dis_kernel<<<gN, B, 0, stream>>>(deg, dis, N);
  wn_kernel<<<gE, B, 0, stream>>>(row, col, ew, dis, wn, E);

  // 2) Chebyshev basis of x:  T1 = L@x ; Tk = 2*L@T(k-1) - T(k-2)
  zero_kernel<<<gF, B, 0, stream>>>(T1, nf);
  spmm32_kernel<<<gEF, B, 0, stream>>>(row, col, wn, x, T1, 1.0f, E);
  scale_copy_kernel<<<gF, B, 0, stream>>>(T2, x, -1.0f, nf);
  spmm32_kernel<<<gEF, B, 0, stream>>>(row, col, wn, T1, T2, 2.0f, E);
  scale_copy_kernel<<<gF, B, 0, stream>>>(T3, T1, -1.0f, nf);
  spmm32_kernel<<<gEF, B, 0, stream>>>(row, col, wn, T2, T3, 2.0f, E);
  scale_copy_kernel<<<gF, B, 0, stream>>>(T4, T2, -1.0f, nf);
  spmm32_kernel<<<gEF, B, 0, stream>>>(row, col, wn, T3, T4, 2.0f, E);

  // 3) cell 1 (H=0): h1 = relu((1-sigmoid(chebZ+bh)) * tanh(chebH+bh))
  gemm_cell1_kernel<<<gT, B, 0, stream>>>(x, T1, T2, T3, T4, Wx1, bx1, bh1, h1, nTiles);

  // 4) Chebyshev basis of h1 (reuse T buffers)
  zero_kernel<<<gF, B, 0, stream>>>(T1, nf);
  spmm32_kernel<<<gEF, B, 0, stream>>>(row, col, wn, h1, T1, 1.0f, E);
  scale_copy_kernel<<<gF, B, 0, stream>>>(T2, h1, -1.0f, nf);
  spmm32_kernel<<<gEF, B, 0, stream>>>(row, col, wn, T1, T2, 2.0f, E);
  scale_copy_kernel<<<gF, B, 0, stream>>>(T3, T1, -1.0f, nf);
  spmm32_kernel<<<gEF, B, 0, stream>>>(row, col, wn, T2, T3, 2.0f, E);
  scale_copy_kernel<<<gF, B, 0, stream>>>(T4, T2, -1.0f, nf);
  spmm32_kernel<<<gEF, B, 0, stream>>>(row, col, wn, T3, T4, 2.0f, E);

  // 5) cell 2 (H=0)
  gemm_cell2_kernel<<<gT, B, 0, stream>>>(h1, T1, T2, T3, T4, Wx2, bx2, bh2, h2, nTiles);

  // 6) output linear
  gemm_out_kernel<<<gT, B, 0, stream>>>(h2, Wl, bl, out, nTiles);
}